// HMM_42966852829305
// MI455X (gfx1250) — compile-verified
//
#include <hip/hip_runtime.h>

typedef float v2f __attribute__((ext_vector_type(2)));
typedef float v8f __attribute__((ext_vector_type(8)));

#define T_LEN 1024
#define S_LEN 512
#define D_LEN 32
#define K_LEN 64
#define LOG2PI_F 1.8378770664093453f

// workspace layout (in floats)
#define WS_BCT   0      // 64*64  : Bc transposed  (BcT[k][j])
#define WS_TT    4096   // 64*64  : trans transposed (TT[k'][k])
#define WS_BIAS  8192   // 64     : per-state bias
#define WS_PART  8256   // 32     : per-block partial log-denominator sums

#define LROW 66         // padded LDS row stride (floats) to avoid bank conflicts

// ---------------------------------------------------------------------------
// Kernel 1: precompute emission coefficient matrix (transposed), bias, trans^T
// ---------------------------------------------------------------------------
__global__ __launch_bounds__(64) void hmm_precompute(
    const float* __restrict__ trans,   // [K,K]
    const float* __restrict__ means,   // [K,D]
    const float* __restrict__ covars,  // [K,D]
    float* __restrict__ ws)
{
    int k = threadIdx.x;               // 0..63 (one state per thread)
    float bias = 0.0f;
    #pragma unroll 4
    for (int j = 0; j < D_LEN; ++j) {
        float cv  = covars[k * D_LEN + j];
        float inv = 1.0f / cv;
        float mn  = means[k * D_LEN + j];
        ws[WS_BCT + k * K_LEN + j]         = -0.5f * inv;   // x^2 coefficient
        ws[WS_BCT + k * K_LEN + D_LEN + j] = mn * inv;      // x coefficient
        bias += mn * mn * inv + __logf(cv);
    }
    ws[WS_BIAS + k] = -0.5f * (bias + (float)D_LEN * LOG2PI_F);
    for (int b = 0; b < K_LEN; ++b)
        ws[WS_TT + k * K_LEN + b] = trans[b * K_LEN + k];   // TT[k'][k] = A[k][k']
}

// ---------------------------------------------------------------------------
// Kernel 2: fused emission + forward recurrence.
// grid = S/16 blocks, 1 wave (32 threads) per block; each wave owns 16 seqs.
// alpha tile [16,64] held as 4 WMMA f32 C/D fragments; both GEMMs use
// V_WMMA_F32_16X16X4_F32 with operands fetched as v2f pairs from padded LDS.
// ---------------------------------------------------------------------------
__global__ __launch_bounds__(32) void hmm_forward(
    const float* __restrict__ data,    // [T,S,D]
    const float* __restrict__ initp,   // [K]
    const float* __restrict__ ws,      // constants from kernel 1
    float* __restrict__ out,           // [S*K + 1]
    float* __restrict__ partials)      // [32]
{
    __shared__ float sBcT[K_LEN * LROW];    // BcT[k][j], padded rows
    __shared__ float sTT [K_LEN * LROW];    // TT[k'][k], padded rows
    __shared__ float sBias[K_LEN];
    __shared__ float sInit[K_LEN];
    __shared__ float sAlpha[16 * LROW];     // alpha[m][k], padded rows

    const int lane = threadIdx.x;      // 0..31
    const int ln   = lane & 15;        // column / A-row index
    const int h    = lane >> 4;        // half: selects k-pair offset 2h
    const int s0   = blockIdx.x * 16;  // first sequence of this tile

    // stage constants into LDS (with row padding)
    for (int i = lane; i < K_LEN * K_LEN; i += 32) {
        int r = i >> 6, c = i & 63;
        sBcT[r * LROW + c] = ws[WS_BCT + i];
        sTT [r * LROW + c] = ws[WS_TT  + i];
    }
    for (int i = lane; i < K_LEN; i += 32) {
        sBias[i] = ws[WS_BIAS + i];
        sInit[i] = initp[i];
    }
    __syncthreads();

    float biasv[4], initv[4];
    #pragma unroll
    for (int n = 0; n < 4; ++n) {
        biasv[n] = sBias[16 * n + ln];
        initv[n] = sInit[16 * n + ln];
    }

    float logden[8];
    #pragma unroll
    for (int v = 0; v < 8; ++v) logden[v] = 0.0f;

    // prefetch t=0 data: lane (h,ln) needs x[row=ln][4c+2h .. +1], c=0..7
    v2f xb[8];
    {
        const float* row = data + ((size_t)0 * S_LEN + (s0 + ln)) * D_LEN;
        #pragma unroll
        for (int c = 0; c < 8; ++c)
            xb[c] = *(const v2f*)(row + 4 * c + 2 * h);
    }

    for (int t = 0; t < T_LEN; ++t) {
        v2f xcur[8];
        #pragma unroll
        for (int c = 0; c < 8; ++c) xcur[c] = xb[c];
        if (t + 1 < T_LEN) {   // software prefetch next timestep
            const float* row = data + ((size_t)(t + 1) * S_LEN + (s0 + ln)) * D_LEN;
            #pragma unroll
            for (int c = 0; c < 8; ++c)
                xb[c] = *(const v2f*)(row + 4 * c + 2 * h);
        }

        // ---- GEMM 1: emission log-prob = [x^2 | x] @ Bc  (K-dim = 64, 16 chunks)
        v8f accP[4];
        #pragma unroll
        for (int n = 0; n < 4; ++n) {
            v8f acc = {};
            const float* brow = &sBcT[(16 * n + ln) * LROW + 2 * h];
            #pragma unroll
            for (int c = 0; c < 8; ++c) {               // squared chunks (j = 0..31)
                v2f a;
                a[0] = xcur[c][0] * xcur[c][0];
                a[1] = xcur[c][1] * xcur[c][1];
                v2f b = *(const v2f*)(brow + 4 * c);
                acc = __builtin_amdgcn_wmma_f32_16x16x4_f32(
                        false, a, false, b, (short)0, acc, false, false);
            }
            #pragma unroll
            for (int c = 0; c < 8; ++c) {               // linear chunks (j = 32..63)
                v2f b = *(const v2f*)(brow + D_LEN + 4 * c);
                acc = __builtin_amdgcn_wmma_f32_16x16x4_f32(
                        false, xcur[c], false, b, (short)0, acc, false, false);
            }
            accP[n] = acc;
        }

        // p = exp(logp + bias)
        float p[4][8];
        #pragma unroll
        for (int n = 0; n < 4; ++n)
            #pragma unroll
            for (int v = 0; v < 8; ++v)
                p[n][v] = __expf(accP[n][v] + biasv[n]);

        // ---- GEMM 2: alpha @ A  (t>0); t==0 uses initial probs instead
        float a_val[4][8];
        if (t == 0) {
            #pragma unroll
            for (int n = 0; n < 4; ++n)
                #pragma unroll
                for (int v = 0; v < 8; ++v)
                    a_val[n][v] = p[n][v] * initv[n];
        } else {
            v2f af[16];
            const float* arow = &sAlpha[ln * LROW + 2 * h];
            #pragma unroll
            for (int c = 0; c < 16; ++c)
                af[c] = *(const v2f*)(arow + 4 * c);
            #pragma unroll
            for (int n = 0; n < 4; ++n) {
                v8f acc = {};
                const float* brow = &sTT[(16 * n + ln) * LROW + 2 * h];
                #pragma unroll
                for (int c = 0; c < 16; ++c) {
                    v2f b = *(const v2f*)(brow + 4 * c);
                    acc = __builtin_amdgcn_wmma_f32_16x16x4_f32(
                            false, af[c], false, b, (short)0, acc, false, false);
                }
                #pragma unroll
                for (int v = 0; v < 8; ++v)
                    a_val[n][v] = p[n][v] * acc[v];
            }
        }

        // ---- row sums over K=64: 4 tile partials, then 16-lane butterfly per half
        float d8[8];
        #pragma unroll
        for (int v = 0; v < 8; ++v)
            d8[v] = a_val[0][v] + a_val[1][v] + a_val[2][v] + a_val[3][v];
        #pragma unroll
        for (int mask = 1; mask <= 8; mask <<= 1)
            #pragma unroll
            for (int v = 0; v < 8; ++v)
                d8[v] += __shfl_xor(d8[v], mask, 32);

        float r8[8];
        #pragma unroll
        for (int v = 0; v < 8; ++v) {
            r8[v] = 1.0f / d8[v];
            logden[v] += __logf(d8[v]);
        }

        // normalize and stage alpha for next step (C/D frag -> row-major LDS)
        #pragma unroll
        for (int n = 0; n < 4; ++n)
            #pragma unroll
            for (int v = 0; v < 8; ++v)
                sAlpha[(v + 8 * h) * LROW + 16 * n + ln] = a_val[n][v] * r8[v];
        __syncthreads();
    }

    // ---- write final alpha [S,K]
    #pragma unroll
    for (int n = 0; n < 4; ++n)
        #pragma unroll
        for (int v = 0; v < 8; ++v)
            out[(size_t)(s0 + v + 8 * h) * K_LEN + 16 * n + ln] =
                sAlpha[(v + 8 * h) * LROW + 16 * n + ln];

    // ---- per-block partial of sum_s log_denom_s (logden replicated per half)
    float lsum = 0.0f;
    #pragma unroll
    for (int v = 0; v < 8; ++v) lsum += logden[v];   // rows of this half
    lsum += __shfl_xor(lsum, 16, 32);                // add other half's rows
    if (lane == 0) partials[blockIdx.x] = lsum;
}

// ---------------------------------------------------------------------------
// Kernel 3: deterministic final reduction  -> out[S*K] = -sum(partials)
// ---------------------------------------------------------------------------
__global__ __launch_bounds__(32) void hmm_reduce(
    const float* __restrict__ partials, float* __restrict__ out)
{
    if (threadIdx.x == 0) {
        float s = 0.0f;
        for (int i = 0; i < 32; ++i) s += partials[i];
        out[S_LEN * K_LEN] = -s;
    }
}

// ---------------------------------------------------------------------------
extern "C" void kernel_launch(void* const* d_in, const int* in_sizes, int n_in,
                              void* d_out, int out_size, void* d_ws, size_t ws_size,
                              hipStream_t stream) {
    const float* data   = (const float*)d_in[0];   // [T,S,D]
    const float* initp  = (const float*)d_in[1];   // [K]
    const float* trans  = (const float*)d_in[2];   // [K,K]
    const float* means  = (const float*)d_in[3];   // [K,D]
    const float* covars = (const float*)d_in[4];   // [K,D]
    float* ws  = (float*)d_ws;
    float* out = (float*)d_out;

    hmm_precompute<<<1, 64, 0, stream>>>(trans, means, covars, ws);
    hmm_forward<<<S_LEN / 16, 32, 0, stream>>>(data, initp, ws, out, ws + WS_PART);
    hmm_reduce<<<1, 32, 0, stream>>>(ws + WS_PART, out);
}